// RecurrentNetwork_15848429323005
// MI455X (gfx1250) — compile-verified
//
#include <hip/hip_runtime.h>
#include <stdint.h>

#define B_    1024
#define NI_   64
#define NN_   8256
#define E_    131072
#define NOUT_ 64
#define NV_   (NI_ + NN_)   // 8320 rows in the value table

#define BT_   256           // threads per block in pass kernel
#define VEC_  2             // batch columns per thread (float2)
#define ROWS_ 8             // dst rows per block
#define MAXE_ 2048          // LDS-staged CSR entries per block (mean ~127)

// ---------------- CDNA5 async-to-LDS helpers (gfx1250) ----------------
// Per-lane 8-byte gather from global into LDS, tracked by ASYNCcnt.
__device__ __forceinline__ void async_gather_b64(uint32_t lds_off, const void* gaddr) {
  asm volatile("global_load_async_to_lds_b64 %0, %1, off"
               :: "v"(lds_off), "v"(gaddr) : "memory");
}
__device__ __forceinline__ void wait_async_0() {
  asm volatile("s_wait_asynccnt 0" ::: "memory");
}

// ---------------- prep kernels ----------------
__global__ void k_zero(float4* __restrict__ v0_state, int n4, int* __restrict__ counts) {
  const int stride = gridDim.x * blockDim.x;
  for (int i = blockIdx.x * blockDim.x + threadIdx.x; i < n4; i += stride)
    v0_state[i] = make_float4(0.f, 0.f, 0.f, 0.f);
  for (int i = blockIdx.x * blockDim.x + threadIdx.x; i < NN_; i += stride)
    counts[i] = 0;
}

// inputs [B, NI] row-major -> V[s*B + b] for s<NI, written into both ping-pong tables
__global__ void k_transpose(const float* __restrict__ in,
                            float* __restrict__ v0, float* __restrict__ v1) {
  const int idx = blockIdx.x * blockDim.x + threadIdx.x;  // idx = s*B + b
  const int s = idx >> 10;
  const int b = idx & (B_ - 1);
  const float v = in[b * NI_ + s];
  v0[idx] = v;
  v1[idx] = v;
}

__global__ void k_count(const int* __restrict__ dst, int* __restrict__ counts) {
  const int e = blockIdx.x * 256 + threadIdx.x;
  atomicAdd(&counts[dst[e]], 1);
}

// exclusive scan of counts[NN] -> row_start[NN+1]; also seed cursor
__global__ __launch_bounds__(1024) void k_scan(const int* __restrict__ counts,
                                               int* __restrict__ row_start,
                                               int* __restrict__ cursor) {
  __shared__ int csum[1024];
  const int t = threadIdx.x;
  const int CH = 9;                 // 1024*9 = 9216 >= 8256
  const int base = t * CH;
  int loc[CH];
  int s = 0;
  for (int i = 0; i < CH; ++i) {
    const int idx = base + i;
    const int c = (idx < NN_) ? counts[idx] : 0;
    loc[i] = s;
    s += c;
  }
  csum[t] = s;
  const int myTot = s;
  __syncthreads();
  for (int off = 1; off < 1024; off <<= 1) {
    const int v = (t >= off) ? csum[t - off] : 0;
    __syncthreads();
    csum[t] += v;
    __syncthreads();
  }
  const int chunkOff = csum[t] - myTot;   // exclusive offset of this chunk
  for (int i = 0; i < CH; ++i) {
    const int idx = base + i;
    if (idx < NN_) {
      const int v = chunkOff + loc[i];
      row_start[idx] = v;
      cursor[idx] = v;
    }
  }
  if (t == 1023) row_start[NN_] = csum[1023];  // == E_
}

// interleave {src, weight-bits} so the hot kernel stages one b64 per edge
__global__ void k_scatter(const int* __restrict__ src, const int* __restrict__ dst,
                          const float* __restrict__ w, int* __restrict__ cursor,
                          int2* __restrict__ csr_ent) {
  const int e = blockIdx.x * 256 + threadIdx.x;
  __builtin_prefetch(src + e + 4096, 0, 0);
  const int d = dst[e];
  const int pos = atomicAdd(&cursor[d], 1);
  csr_ent[pos] = make_int2(src[e], __float_as_int(w[e]));
}

// ---------------- hot kernel: one recurrent pass ----------------
// Vin:  [NV_, B] value table (inputs rows + previous state rows)
// SOut: state region of the other table ( = Vother + NI_*B )
__global__ __launch_bounds__(BT_) void k_pass(const float* __restrict__ Vin,
                                              float* __restrict__ SOut,
                                              const int* __restrict__ row_start,
                                              const int2* __restrict__ csr_ent,
                                              const float* __restrict__ bias,
                                              const float* __restrict__ resp) {
  __shared__ int2 sEnt[MAXE_];

  const int b2   = (blockIdx.x * BT_ + threadIdx.x) * VEC_;  // first of 2 batch cols
  const int row0 = blockIdx.y * ROWS_;                       // first dst row
  const int e0   = row_start[row0];
  const int e1   = row_start[row0 + ROWS_];
  const int cnt  = e1 - e0;
  const bool lds_ok = (cnt <= MAXE_);

  if (lds_ok) {
    // Stage this block's CSR slice into LDS with the CDNA5 async gather path.
    for (int i = threadIdx.x; i < cnt; i += BT_)
      async_gather_b64((uint32_t)(uintptr_t)&sEnt[i], (const void*)(csr_ent + e0 + i));
    wait_async_0();
  }
  __syncthreads();

  for (int r = 0; r < ROWS_; ++r) {
    const int n  = row0 + r;
    const int rs = row_start[n];
    const int re = row_start[n + 1];
    float a0x = 0.f, a0y = 0.f, a1x = 0.f, a1y = 0.f;
    if (lds_ok) {
      int i = rs - e0;
      const int iend = re - e0;
      for (; i + 1 < iend; i += 2) {
        const int2 eA = sEnt[i];
        const int2 eB = sEnt[i + 1];
        const float2 vA = *(const float2*)(Vin + (size_t)eA.x * B_ + b2);
        const float2 vB = *(const float2*)(Vin + (size_t)eB.x * B_ + b2);
        const float wA = __int_as_float(eA.y);
        const float wB = __int_as_float(eB.y);
        a0x = fmaf(wA, vA.x, a0x);
        a0y = fmaf(wA, vA.y, a0y);
        a1x = fmaf(wB, vB.x, a1x);
        a1y = fmaf(wB, vB.y, a1y);
      }
      if (i < iend) {
        const int2 eA = sEnt[i];
        const float2 vA = *(const float2*)(Vin + (size_t)eA.x * B_ + b2);
        const float wA = __int_as_float(eA.y);
        a0x = fmaf(wA, vA.x, a0x);
        a0y = fmaf(wA, vA.y, a0y);
      }
    } else {  // statistically never taken; correctness fallback
      for (int e = rs; e < re; ++e) {
        __builtin_prefetch(csr_ent + e + 16, 0, 0);
        const int2 eA = csr_ent[e];
        const float2 vA = *(const float2*)(Vin + (size_t)eA.x * B_ + b2);
        const float wA = __int_as_float(eA.y);
        a0x = fmaf(wA, vA.x, a0x);
        a0y = fmaf(wA, vA.y, a0y);
      }
    }
    float2 o = make_float2(0.f, 0.f);
    if (re > rs) {  // neurons with no incoming connections output exactly 0
      const float rn = resp[n];
      const float bn = bias[n];
      o.x = tanhf(fmaf(rn, a0x + a1x, bn));
      o.y = tanhf(fmaf(rn, a0y + a1y, bn));
    }
    *(float2*)(SOut + (size_t)n * B_ + b2) = o;
  }
}

// out[b, n] = S[n, b] for n < 64
__global__ void k_out(const float* __restrict__ S, float* __restrict__ out) {
  const int idx = blockIdx.x * 256 + threadIdx.x;   // idx = b*64 + n
  const int n = idx & (NOUT_ - 1);
  const int b = idx >> 6;
  out[idx] = S[(size_t)n * B_ + b];
}

// ---------------- launch ----------------
extern "C" void kernel_launch(void* const* d_in, const int* in_sizes, int n_in,
                              void* d_out, int out_size, void* d_ws, size_t ws_size,
                              hipStream_t stream) {
  const float* inputs  = (const float*)d_in[0];
  const float* weights = (const float*)d_in[1];
  const float* bias    = (const float*)d_in[2];
  const float* resp    = (const float*)d_in[3];
  const int*   src     = (const int*)d_in[4];
  const int*   dst     = (const int*)d_in[5];
  float* out = (float*)d_out;

  // workspace layout (regions padded to 256B multiples)
  float* V0        = (float*)d_ws;                      // NV_*B_ floats (34 MB)
  float* V1        = V0 + (size_t)NV_ * B_;             // NV_*B_ floats (34 MB)
  int*   row_start = (int*)(V1 + (size_t)NV_ * B_);     // NN_+1 -> padded 8320
  int*   counts    = row_start + 8320;                  // NN_   -> padded 8320
  int*   cursor    = counts + 8320;                     // NN_   -> padded 8320
  int2*  csr_ent   = (int2*)(cursor + 8320);            // E_ entries (1 MB)

  float* S0 = V0 + (size_t)NI_ * B_;   // state region of V0
  float* S1 = V1 + (size_t)NI_ * B_;   // state region of V1

  // 1) zero V0 state rows (state0 = 0) and the dst histogram
  k_zero<<<1024, 256, 0, stream>>>((float4*)S0, (NN_ * B_) / 4, counts);
  // 2) inputs^T into both ping-pong tables
  k_transpose<<<(NI_ * B_) / 256, 256, 0, stream>>>(inputs, V0, V1);
  // 3) build CSR: histogram -> scan -> scatter (interleaved {src, w})
  k_count<<<E_ / 256, 256, 0, stream>>>(dst, counts);
  k_scan<<<1, 1024, 0, stream>>>(counts, row_start, cursor);
  k_scatter<<<E_ / 256, 256, 0, stream>>>(src, dst, weights, cursor, csr_ent);
  // 4) two recurrent passes (V0 -> V1 -> V0)
  dim3 grid(B_ / (BT_ * VEC_), NN_ / ROWS_);
  k_pass<<<grid, BT_, 0, stream>>>(V0, S1, row_start, csr_ent, bias, resp);
  k_pass<<<grid, BT_, 0, stream>>>(V1, S0, row_start, csr_ent, bias, resp);
  // 5) emit [B, 64]
  k_out<<<(B_ * NOUT_) / 256, 256, 0, stream>>>(S0, out);
}